// GNNModel_22643067584885
// MI455X (gfx1250) — compile-verified
//
#include <hip/hip_runtime.h>
#include <hip/hip_bf16.h>
#include <math.h>

typedef __attribute__((ext_vector_type(2))) float v2f;
typedef __attribute__((ext_vector_type(8))) float v8f;

#define NUM_USERS 50000
#define N_NODES   100000
#define N_EDGES   3200000
#define BATCH     16384

// ---------------------------------------------------------------------------
// Dense GEMM  out[N,M] = A[N,K] @ W[K,M]  via V_WMMA_F32_16X16X4_F32.
// One wave32 per 16x16 output tile, K-loop in steps of 4.
// VGPR layouts per CDNA5 ISA 7.12.2:
//   A 16x4 : lane L -> row L&15, vgpr v -> k = (L>>4)*2 + v
//   B 4x16 : lane L -> col L&15, vgpr v -> k = (L>>4)*2 + v
//   C/D    : vgpr r -> row r + 8*(L>=16), col L&15
// N (=100000) is an exact multiple of 16, M is 128 or 64.
// ---------------------------------------------------------------------------
template<int K, int M>
__global__ __launch_bounds__(256)
void gemm_wmma_f32(const float* __restrict__ A, const float* __restrict__ W,
                   float* __restrict__ out, int N) {
  const int lane = threadIdx.x & 31;
  const int wave = blockIdx.x * (blockDim.x >> 5) + (threadIdx.x >> 5);
  constexpr int tilesM = M / 16;
  const int tN = wave / tilesM;
  const int tM = wave - tN * tilesM;
  if (tN * 16 >= N) return;                 // whole-wave exit: EXEC all-1 for WMMA

  const int r16  = lane & 15;               // A row / B,C,D col within tile
  const int koff = (lane >> 4) << 1;        // 0 or 2

  const float* Arow = A + (size_t)(tN * 16 + r16) * K;
  const float* Bcol = W + (size_t)(tM * 16 + r16);

  v8f acc = {};
#pragma unroll
  for (int k = 0; k < K; k += 4) {
    v2f a, b;
    a[0] = Arow[k + koff + 0];
    a[1] = Arow[k + koff + 1];
    b[0] = Bcol[(size_t)(k + koff + 0) * M];
    b[1] = Bcol[(size_t)(k + koff + 1) * M];
    // 8 args: (neg_a, A, neg_b, B, c_mod, C, reuse_a, reuse_b)
    acc = __builtin_amdgcn_wmma_f32_16x16x4_f32(false, a, false, b,
                                                (short)0, acc, false, false);
  }

  const int rbase = tN * 16 + ((lane >> 4) << 3);
  const int col   = tM * 16 + r16;
#pragma unroll
  for (int r = 0; r < 8; ++r)
    out[(size_t)(rbase + r) * M + col] = acc[r];
}

// ---------------------------------------------------------------------------
// SpMM scatter: agg[src[e], :] += w[e] * sup[dst[e], :]
// D/4 threads per edge; each thread moves one float4 and issues 4 native
// f32 atomic adds (L2-resident: 51.2 MB table << 192 MB L2).
// CSHIFT = log2(D/4).
// ---------------------------------------------------------------------------
template<int D, int CSHIFT>
__global__ __launch_bounds__(256)
void spmm_scatter(const float* __restrict__ sup, const int* __restrict__ esrc,
                  const int* __restrict__ edst, const float* __restrict__ ew,
                  float* __restrict__ agg, int nE) {
  const long long t = (long long)blockIdx.x * blockDim.x + threadIdx.x;
  const int e = (int)(t >> CSHIFT);
  if (e >= nE) return;
  const int c = ((int)t & ((1 << CSHIFT) - 1)) << 2;   // float offset in row
  const float w = ew[e];
  const int j = edst[e];
  const int i = esrc[e];
  const float4 m = *(const float4*)(sup + (size_t)j * D + c);
  float* o = agg + (size_t)i * D + c;
  unsafeAtomicAdd(o + 0, m.x * w);
  unsafeAtomicAdd(o + 1, m.y * w);
  unsafeAtomicAdd(o + 2, m.z * w);
  unsafeAtomicAdd(o + 3, m.w * w);
}

// ---------------------------------------------------------------------------
// x = relu(x + b[col]);  dmask = D-1 (D power of two)
// ---------------------------------------------------------------------------
__global__ __launch_bounds__(256)
void bias_relu(float* __restrict__ x, const float* __restrict__ b,
               int n, int dmask) {
  const int t = blockIdx.x * blockDim.x + threadIdx.x;
  if (t >= n) return;
  const float v = x[t] + b[t & dmask];
  x[t] = v > 0.0f ? v : 0.0f;
}

// ---------------------------------------------------------------------------
// scores[b] = sigmoid(dot64(h[user[b]], h[item[b]+NUM_USERS]))
// ---------------------------------------------------------------------------
__global__ __launch_bounds__(256)
void predict_dot(const float* __restrict__ h, const int* __restrict__ uidx,
                 const int* __restrict__ iidx, float* __restrict__ out, int B) {
  const int t = blockIdx.x * blockDim.x + threadIdx.x;
  if (t >= B) return;
  const float4* u = (const float4*)(h + (size_t)uidx[t] * 64);
  const float4* v = (const float4*)(h + (size_t)(iidx[t] + NUM_USERS) * 64);
  float s = 0.0f;
#pragma unroll
  for (int k = 0; k < 16; ++k) {
    const float4 a = u[k], b = v[k];
    s += a.x * b.x + a.y * b.y + a.z * b.z + a.w * b.w;
  }
  out[t] = 1.0f / (1.0f + expf(-s));
}

// ---------------------------------------------------------------------------
extern "C" void kernel_launch(void* const* d_in, const int* in_sizes, int n_in,
                              void* d_out, int out_size, void* d_ws, size_t ws_size,
                              hipStream_t stream) {
  const float* node_emb = (const float*)d_in[0];   // [100000,64]
  const float* W1       = (const float*)d_in[1];   // [64,128]
  const float* b1       = (const float*)d_in[2];   // [128]
  const float* W2       = (const float*)d_in[3];   // [128,64]
  const float* b2       = (const float*)d_in[4];   // [64]
  const float* ew       = (const float*)d_in[5];   // [E]
  const int*   esrc     = (const int*)d_in[6];     // [E]
  const int*   edst     = (const int*)d_in[7];     // [E]
  const int*   uidx     = (const int*)d_in[8];     // [B]
  const int*   iidx     = (const int*)d_in[9];     // [B]
  float* out = (float*)d_out;

  // workspace: two ping-pong buffers of 100000*128 floats (51.2 MB each)
  float* buf0 = (float*)d_ws;
  float* buf1 = buf0 + (size_t)N_NODES * 128;

  // ---- layer 1: support1 = X @ W1  (K=64, M=128) -> buf0 ----
  {
    const int waves = (N_NODES / 16) * (128 / 16);   // 50000, /8 waves per block
    gemm_wmma_f32<64, 128><<<waves / 8, 256, 0, stream>>>(node_emb, W1, buf0, N_NODES);
  }
  hipMemsetAsync(buf1, 0, (size_t)N_NODES * 128 * sizeof(float), stream);
  {
    const long long th = (long long)N_EDGES * 32;    // D/4 = 32 threads per edge
    spmm_scatter<128, 5><<<(int)((th + 255) / 256), 256, 0, stream>>>(
        buf0, esrc, edst, ew, buf1, N_EDGES);
  }
  {
    const int n = N_NODES * 128;
    bias_relu<<<(n + 255) / 256, 256, 0, stream>>>(buf1, b1, n, 127);
  }

  // ---- layer 2: support2 = h1 @ W2  (K=128, M=64) -> buf0 (reuse) ----
  {
    const int waves = (N_NODES / 16) * (64 / 16);    // 25000
    gemm_wmma_f32<128, 64><<<waves / 8, 256, 0, stream>>>(buf1, W2, buf0, N_NODES);
  }
  hipMemsetAsync(buf1, 0, (size_t)N_NODES * 64 * sizeof(float), stream);
  {
    const long long th = (long long)N_EDGES * 16;    // D/4 = 16 threads per edge
    spmm_scatter<64, 4><<<(int)((th + 255) / 256), 256, 0, stream>>>(
        buf0, esrc, edst, ew, buf1, N_EDGES);
  }
  {
    const int n = N_NODES * 64;
    bias_relu<<<(n + 255) / 256, 256, 0, stream>>>(buf1, b2, n, 63);
  }

  // ---- prediction head ----
  predict_dot<<<(BATCH + 255) / 256, 256, 0, stream>>>(buf1, uidx, iidx, out, BATCH);
}